// MatrixTree_4733053960796
// MI455X (gfx1250) — compile-verified
//
#include <hip/hip_runtime.h>
#include <math.h>

// Problem shape (from reference): B=64, N=512.
#define NDIM 512
#define BDIM 64
#define TPB  256
#define NB   16          // WMMA tile / LU panel width
#define NTIL (NDIM / NB) // 32 tiles per dimension

typedef float v2f __attribute__((ext_vector_type(2)));
typedef float v8f __attribute__((ext_vector_type(8)));

static __device__ __forceinline__ unsigned laneid() {
  return __builtin_amdgcn_mbcnt_lo(~0u, 0u); // wave32: lane id 0..31
}

// ---- WMMA f32 16x16x4 fragment helpers (layouts per CDNA5 ISA 7.12.2) ----
// A (16x4, k-quad q of a 16x16 tile): lanes 0-15 -> M=lane, K={4q,4q+1};
//                                     lanes 16-31 -> M=lane-16, K={4q+2,4q+3}
static __device__ __forceinline__ v2f load_a_frag(const float* t, int ld, int q,
                                                  unsigned lane, bool neg) {
  int m = lane & 15;
  int k = q * 4 + ((lane >> 4) << 1);
  v2f a;
  a.x = t[m * ld + k];
  a.y = t[m * ld + k + 1];
  if (neg) { a.x = -a.x; a.y = -a.y; }
  return a;
}
// B (4x16): lanes 0-15 -> N=lane, K={4q,4q+1} across the 2 VGPRs; lanes 16-31 K+=2
static __device__ __forceinline__ v2f load_b_frag(const float* t, int ld, int q,
                                                  unsigned lane) {
  int n = lane & 15;
  int k = q * 4 + ((lane >> 4) << 1);
  v2f b;
  b.x = t[k * ld + n];
  b.y = t[(k + 1) * ld + n];
  return b;
}
// C/D (16x16, 8 VGPRs): VGPR v -> row v (lanes 0-15) or v+8 (lanes 16-31), col=lane&15
static __device__ __forceinline__ v8f load_c_frag(const float* t, int ld,
                                                  unsigned lane) {
  int n = lane & 15;
  int r0 = (lane >> 4) << 3;
  v8f c;
#pragma unroll
  for (int v = 0; v < 8; ++v) c[v] = t[(r0 + v) * ld + n];
  return c;
}
static __device__ __forceinline__ void store_c_frag(float* t, int ld,
                                                    unsigned lane, v8f c) {
  int n = lane & 15;
  int r0 = (lane >> 4) << 3;
#pragma unroll
  for (int v = 0; v < 8; ++v) t[(r0 + v) * ld + n] = c[v];
}

static __device__ __forceinline__ v8f zero8() {
  v8f z = {0.f, 0.f, 0.f, 0.f, 0.f, 0.f, 0.f, 0.f};
  return z;
}

// C += (negA ? -A : A) * B for one 16x16 tile, as 4 chained k=4 WMMAs.
static __device__ __forceinline__ v8f wmma16(const float* At, int lda,
                                             const float* Bt, int ldb, v8f c,
                                             bool negA) {
  unsigned lane = laneid();
#pragma unroll
  for (int q = 0; q < 4; ++q) {
    v2f a = load_a_frag(At, lda, q, lane, negA);
    v2f b = load_b_frag(Bt, ldb, q, lane);
    c = __builtin_amdgcn_wmma_f32_16x16x4_f32(false, a, false, b, (short)0, c,
                                              false, false);
  }
  return c;
}

// ---------------------------------------------------------------------------
// K0: zero the scalar accumulators: scal[0]=logZ, scal[1]=gold, scal[2]=count
__global__ void mt_zero_kernel(float* scal) {
  if (threadIdx.x < 4) scal[threadIdx.x] = 0.f;
}

// ---------------------------------------------------------------------------
// K1: build padded Laplacian minor M (512x512, decoupled identity row/col 511)
// and accumulate gold score + mask count.
__global__ void mt_build_kernel(const float* __restrict__ scores,
                                const unsigned char* __restrict__ mask,
                                const int* __restrict__ target,
                                float* __restrict__ Mb, float* scal) {
  const int i = blockIdx.x;           // row 0..511
  const int b = blockIdx.y;           // batch
  const int tid = threadIdx.x;        // 256 threads, 2 cols each
  const float* srow = scores + ((size_t)b * NDIM + i) * NDIM;
  float* M = Mb + (size_t)b * NDIM * NDIM;
  const bool mi = (i == 0) ? true : (mask[(size_t)b * NDIM + i] != 0);

  const int j0 = tid, j1 = tid + TPB;
  const bool mj0 = (j0 == 0) ? true : (mask[(size_t)b * NDIM + j0] != 0);
  const bool mj1 = (mask[(size_t)b * NDIM + j1] != 0);
  const float a0 = (mi && mj0) ? __expf(srow[j0]) : 0.f;
  const float a1 = (mi && mj1) ? __expf(srow[j1]) : 0.f;

  __shared__ float red[TPB];
  red[tid] = a0 + a1;
  __syncthreads();
  for (int s = TPB / 2; s > 0; s >>= 1) {
    if (tid < s) red[tid] += red[tid + s];
    __syncthreads();
  }
  const float d = red[0];

  if (i >= 1) {
    const int p = i - 1;
    float* Mrow = M + (size_t)p * NDIM;
    // col from j0
    if (j0 >= 1) {
      const int q = j0 - 1;
      Mrow[q] = mi ? (d * (p == q ? 1.f : 0.f) - a0) : (p == q ? 1.f : 0.f);
    } else {
      Mrow[NDIM - 1] = 0.f; // padded column
    }
    {
      const int q = j1 - 1;
      Mrow[q] = mi ? (d * (p == q ? 1.f : 0.f) - a1) : (p == q ? 1.f : 0.f);
    }
  } else {
    // i==0 blocks write the padded identity row 511
    M[(size_t)(NDIM - 1) * NDIM + j0] = (j0 == NDIM - 1) ? 1.f : 0.f;
    M[(size_t)(NDIM - 1) * NDIM + j1] = (j1 == NDIM - 1) ? 1.f : 0.f;
  }

  if (tid == 0 && mask[(size_t)b * NDIM + i]) {
    atomicAdd(&scal[1], srow[target[(size_t)b * NDIM + i]]);
    atomicAdd(&scal[2], 1.0f);
  }
}

// ---------------------------------------------------------------------------
// K2: blocked right-looking LU (no pivoting; Laplacian minor is an M-matrix),
// trailing rank-16 updates on WMMA. One workgroup (8 waves) per batch.
__global__ void mt_lu_kernel(float* __restrict__ Mb, float* scal) {
  const int b = blockIdx.x;
  float* M = Mb + (size_t)b * NDIM * NDIM;
  const int tid = threadIdx.x;
  const int wave = tid >> 5;
  const unsigned lane = laneid();
  __shared__ float sL11[NB * NB];
  __shared__ float spiv;
  float logdet = 0.f;

  for (int kb = 0; kb < NTIL; ++kb) {
    const int c0 = kb * NB;
    // ---- unblocked panel factorization (columns c0..c0+15) ----
    for (int jj = 0; jj < NB; ++jj) {
      const int col = c0 + jj;
      __syncthreads();
      if (tid == 0) {
        const float piv = M[(size_t)col * NDIM + col];
        spiv = piv;
        logdet += __logf(fabsf(piv));
      }
      __syncthreads();
      const float piv = spiv;
      for (int i = col + 1 + tid; i < NDIM; i += TPB)
        M[(size_t)i * NDIM + col] /= piv;
      __syncthreads();
      for (int i = col + 1 + tid; i < NDIM; i += TPB) {
        const float lik = M[(size_t)i * NDIM + col];
#pragma unroll
        for (int j2 = 0; j2 < NB; ++j2)
          if (j2 > jj)
            M[(size_t)i * NDIM + c0 + j2] -=
                lik * M[(size_t)col * NDIM + c0 + j2];
      }
    }
    __syncthreads();
    // ---- stage unit-lower L11 in LDS ----
    {
      const int r = tid >> 4, cc = tid & 15;
      sL11[r * NB + cc] = (r > cc) ? M[(size_t)(c0 + r) * NDIM + c0 + cc]
                                   : (r == cc ? 1.f : 0.f);
    }
    __syncthreads();
    // ---- U12 = L11^{-1} * A12 : forward substitution, one column/thread ----
    for (int c = c0 + NB + tid; c < NDIM; c += TPB) {
      float x[NB];
#pragma unroll
      for (int r = 0; r < NB; ++r) {
        float v = M[(size_t)(c0 + r) * NDIM + c];
        for (int t = 0; t < r; ++t) v -= sL11[r * NB + t] * x[t];
        x[r] = v;
      }
#pragma unroll
      for (int r = 0; r < NB; ++r) M[(size_t)(c0 + r) * NDIM + c] = x[r];
    }
    __syncthreads();
    // ---- trailing update A22 -= L21 * U12 : WMMA tiles, 8 waves ----
    const int nt = NTIL - 1 - kb;
    const int tot = nt * nt;
    for (int t = wave; t < tot; t += 8) {
      const int ti = kb + 1 + t / nt;
      const int tj = kb + 1 + t % nt;
      float* Ct = M + (size_t)ti * NB * NDIM + tj * NB;
      const float* At = M + (size_t)ti * NB * NDIM + c0;   // L21 tile
      const float* Bt = M + (size_t)c0 * NDIM + tj * NB;   // U12 tile
      v8f c = load_c_frag(Ct, NDIM, lane);
      c = wmma16(At, NDIM, Bt, NDIM, c, /*negA=*/true);
      store_c_frag(Ct, NDIM, lane, c);
    }
    __syncthreads();
  }
  if (tid == 0) atomicAdd(&scal[0], logdet);
}

// ---------------------------------------------------------------------------
// K3: blocked triangular inverses Linv = L^{-1} (unit lower), Uinv = U^{-1}.
// Off-diagonal block recurrences are 16x16x16 WMMA chains.
__global__ void mt_trinv_kernel(const float* __restrict__ Mb,
                                float* __restrict__ Linvb,
                                float* __restrict__ Uinvb) {
  const int b = blockIdx.x;
  const float* M = Mb + (size_t)b * NDIM * NDIM;
  float* Linv = Linvb + (size_t)b * NDIM * NDIM;
  float* Uinv = Uinvb + (size_t)b * NDIM * NDIM;
  const int tid = threadIdx.x;
  const int wave = tid >> 5;
  const unsigned lane = laneid();
  __shared__ float sD[NB * NB];
  __shared__ float sDi[NB * NB];
  __shared__ float sT[8][NB * NB]; // per-wave staging tile

  // ================= Linv (bi ascending) =================
  for (int bi = 0; bi < NTIL; ++bi) {
    {
      const int r = tid >> 4, cc = tid & 15;
      sD[r * NB + cc] = (r > cc) ? M[(size_t)(bi * NB + r) * NDIM + bi * NB + cc]
                                 : (r == cc ? 1.f : 0.f);
    }
    __syncthreads();
    if (tid < NB) { // invert unit-lower diag block, one column per thread
      const int c = tid;
      float x[NB];
#pragma unroll
      for (int r = 0; r < NB; ++r) {
        float v = (r == c) ? 1.f : 0.f;
        for (int t = c; t < r; ++t) v -= sD[r * NB + t] * x[t];
        x[r] = v;
      }
#pragma unroll
      for (int r = 0; r < NB; ++r) sDi[r * NB + c] = (r >= c) ? x[r] : 0.f;
    }
    __syncthreads();
    {
      const int r = tid >> 4, cc = tid & 15;
      Linv[(size_t)(bi * NB + r) * NDIM + bi * NB + cc] = sDi[r * NB + cc];
    }
    for (int bj = wave; bj < bi; bj += 8) {
      v8f acc = zero8();
      for (int k = bj; k < bi; ++k)
        acc = wmma16(M + (size_t)(bi * NB) * NDIM + k * NB, NDIM,
                     Linv + (size_t)(k * NB) * NDIM + bj * NB, NDIM, acc, false);
      store_c_frag(&sT[wave][0], NB, lane, acc);
      asm volatile("s_wait_dscnt 0" ::: "memory"); // wave-local LDS RAW
      v8f r2 = wmma16(sDi, NB, &sT[wave][0], NB, zero8(), /*negA=*/true);
      store_c_frag(Linv + (size_t)(bi * NB) * NDIM + bj * NB, NDIM, lane, r2);
    }
    __syncthreads();
  }

  // ================= Uinv (bi descending) =================
  for (int s = 0; s < NTIL; ++s) {
    const int bi = NTIL - 1 - s;
    {
      const int r = tid >> 4, cc = tid & 15;
      sD[r * NB + cc] =
          (r <= cc) ? M[(size_t)(bi * NB + r) * NDIM + bi * NB + cc] : 0.f;
    }
    __syncthreads();
    if (tid < NB) { // invert upper diag block, back substitution per column
      const int c = tid;
      float x[NB];
      for (int r = c; r >= 0; --r) {
        float v = (r == c) ? 1.f : 0.f;
        for (int t = r + 1; t <= c; ++t) v -= sD[r * NB + t] * x[t];
        x[r] = v / sD[r * NB + r];
      }
#pragma unroll
      for (int r = 0; r < NB; ++r) sDi[r * NB + c] = (r <= c) ? x[r] : 0.f;
    }
    __syncthreads();
    {
      const int r = tid >> 4, cc = tid & 15;
      Uinv[(size_t)(bi * NB + r) * NDIM + bi * NB + cc] = sDi[r * NB + cc];
    }
    for (int bj = bi + 1 + wave; bj < NTIL; bj += 8) {
      v8f acc = zero8();
      for (int k = bi + 1; k <= bj; ++k)
        acc = wmma16(M + (size_t)(bi * NB) * NDIM + k * NB, NDIM,
                     Uinv + (size_t)(k * NB) * NDIM + bj * NB, NDIM, acc, false);
      store_c_frag(&sT[wave][0], NB, lane, acc);
      asm volatile("s_wait_dscnt 0" ::: "memory");
      v8f r2 = wmma16(sDi, NB, &sT[wave][0], NB, zero8(), /*negA=*/true);
      store_c_frag(Uinv + (size_t)(bi * NB) * NDIM + bj * NB, NDIM, lane, r2);
    }
    __syncthreads();
  }
}

// ---------------------------------------------------------------------------
// K4: Minv = Uinv * Linv (triangular x triangular), pure tiled WMMA GEMM.
// One wave (32 threads = full wave32) per 16x16 output tile; overwrites M.
__global__ void mt_minv_kernel(float* __restrict__ Mb,
                               const float* __restrict__ Linvb,
                               const float* __restrict__ Uinvb) {
  const int bi = blockIdx.x, bj = blockIdx.y, b = blockIdx.z;
  float* M = Mb + (size_t)b * NDIM * NDIM;
  const float* Linv = Linvb + (size_t)b * NDIM * NDIM;
  const float* Uinv = Uinvb + (size_t)b * NDIM * NDIM;
  const unsigned lane = laneid();
  const int k0 = (bi > bj) ? bi : bj;
  v8f acc = zero8();
  for (int k = k0; k < NTIL; ++k)
    acc = wmma16(Uinv + (size_t)(bi * NB) * NDIM + k * NB, NDIM,
                 Linv + (size_t)(k * NB) * NDIM + bj * NB, NDIM, acc, false);
  store_c_frag(M + (size_t)(bi * NB) * NDIM + bj * NB, NDIM, lane, acc);
}

// ---------------------------------------------------------------------------
// K5: marginals[b,i,j] = A[i,j]*(Minv[i-1,i-1] - Minv[j-1,i-1]) and the loss.
__global__ void mt_marg_kernel(const float* __restrict__ scores,
                               const unsigned char* __restrict__ mask,
                               const float* __restrict__ Minvb,
                               const float* __restrict__ scal,
                               float* __restrict__ out) {
  const int i = blockIdx.x;
  const int b = blockIdx.y;
  const int tid = threadIdx.x;
  const float* Minv = Minvb + (size_t)b * NDIM * NDIM;
  const float* srow = scores + ((size_t)b * NDIM + i) * NDIM;
  float* mrow = out + 1 + ((size_t)b * NDIM + i) * NDIM;
  const bool mi = (i == 0) ? true : (mask[(size_t)b * NDIM + i] != 0);
  const float diag = (i >= 1) ? Minv[(size_t)(i - 1) * NDIM + (i - 1)] : 0.f;

#pragma unroll
  for (int u = 0; u < 2; ++u) {
    const int j = tid + u * TPB;
    const bool mj = (j == 0) ? true : (mask[(size_t)b * NDIM + j] != 0);
    float v = 0.f;
    if (i >= 1 && mi && mj) {
      const float term =
          diag - ((j >= 1) ? Minv[(size_t)(j - 1) * NDIM + (i - 1)] : 0.f);
      v = __expf(srow[j]) * term;
    }
    mrow[j] = v;
  }
  if (b == 0 && i == 0 && tid == 0)
    out[0] = (scal[0] - scal[1]) / scal[2];
}

// ---------------------------------------------------------------------------
extern "C" void kernel_launch(void* const* d_in, const int* in_sizes, int n_in,
                              void* d_out, int out_size, void* d_ws,
                              size_t ws_size, hipStream_t stream) {
  const float* scores = (const float*)d_in[0];
  const unsigned char* mask = (const unsigned char*)d_in[1];
  const int* target = (const int*)d_in[2];
  float* out = (float*)d_out;

  const size_t matBytes = (size_t)BDIM * NDIM * NDIM * sizeof(float); // 64 MB
  float* Mb = (float*)d_ws;
  float* Linv = (float*)((char*)d_ws + matBytes);
  float* Uinv = (float*)((char*)d_ws + 2 * matBytes);
  float* scal = (float*)((char*)d_ws + 3 * matBytes);

  mt_zero_kernel<<<1, 32, 0, stream>>>(scal);
  mt_build_kernel<<<dim3(NDIM, BDIM), TPB, 0, stream>>>(scores, mask, target,
                                                        Mb, scal);
  mt_lu_kernel<<<BDIM, TPB, 0, stream>>>(Mb, scal);
  mt_trinv_kernel<<<BDIM, TPB, 0, stream>>>(Mb, Linv, Uinv);
  mt_minv_kernel<<<dim3(NTIL, NTIL, BDIM), 32, 0, stream>>>(Mb, Linv, Uinv);
  mt_marg_kernel<<<dim3(NDIM, BDIM), TPB, 0, stream>>>(scores, mask, Mb, scal,
                                                       out);
}